// ImagePairFlowPredictor_33844342292838
// MI455X (gfx1250) — compile-verified
//
#include <hip/hip_runtime.h>

// ---------------------------------------------------------------------------
// ImagePairFlowPredictor for MI455X (gfx1250).
// Memory-bound workload (~4 GFLOP vs ~0.5 GB traffic @ 23.3 TB/s).
// The 512->16 per-patch GEMM runs on V_WMMA_F32_16X16X4_F32 (fp32 WMMA,
// wave32; 16 patches x 16 features per tile, K=512 in 128 steps of 4).
// w2 is staged to LDS via the CDNA5 async-to-LDS path (ASYNCcnt) so the
// WMMA loop is pure DS + WMMA with no VMEM waits on the critical chain.
// ---------------------------------------------------------------------------

#define PP     8
#define EPSV   1e-5f
#define LDA    516          // padded A row stride (floats): 516 % 64 = 4 banks
#define LDH    17

typedef __attribute__((ext_vector_type(2))) float v2f;
typedef __attribute__((ext_vector_type(8))) float v8f;
typedef __attribute__((ext_vector_type(4))) int   v4i;

#if defined(__gfx1250__) && __has_builtin(__builtin_amdgcn_global_load_async_to_lds_b128)
#define USE_ASYNC_LDS 1
#else
#define USE_ASYNC_LDS 0
#endif

__device__ __forceinline__ void wait_async0() {
#if __has_builtin(__builtin_amdgcn_s_wait_asynccnt)
    __builtin_amdgcn_s_wait_asynccnt(0);
#else
    asm volatile("s_wait_asynccnt 0" ::: "memory");
#endif
}

// ---------------------------------------------------------------------------
// db2 wavelet downsample: 4x4 cross-correlation, stride 2, edge padding of 1.
// src: channel-0 plane per batch (stride srcBatchStride floats)
// dst: (B, 4, Ho, Wo) with bands {LL, LH, HL, HH}
// ---------------------------------------------------------------------------
__global__ void wavelet_down(const float* __restrict__ src, float* __restrict__ dst,
                             int B, int Hin, int Win, long srcBatchStride)
{
    const float LO[4] = {-0.12940952255092145f, 0.22414386804185735f,
                          0.836516303737469f,   0.48296291314469025f};
    const float HI[4] = {-0.48296291314469025f, 0.836516303737469f,
                         -0.22414386804185735f, -0.12940952255092145f};
    const int Ho = Hin >> 1, Wo = Win >> 1;
    long idx = (long)blockIdx.x * blockDim.x + threadIdx.x;
    long N = (long)B * Ho * Wo;
    if (idx >= N) return;
    int ox = (int)(idx % Wo);
    long t  = idx / Wo;
    int oy = (int)(t % Ho);
    int b  = (int)(t / Ho);

    const float* s = src + (long)b * srcBatchStride;
    float colL[4], colH[4];
#pragma unroll
    for (int j = 0; j < 4; ++j) {
        int ix = 2 * ox - 1 + j;
        ix = ix < 0 ? 0 : (ix > Win - 1 ? Win - 1 : ix);
        float cl = 0.f, ch = 0.f;
#pragma unroll
        for (int i = 0; i < 4; ++i) {
            int iy = 2 * oy - 1 + i;
            iy = iy < 0 ? 0 : (iy > Hin - 1 ? Hin - 1 : iy);
            float v = s[(long)iy * Win + ix];
            cl = fmaf(v, LO[i], cl);
            ch = fmaf(v, HI[i], ch);
        }
        colL[j] = cl; colH[j] = ch;
    }
    float ll = 0.f, lh = 0.f, hl = 0.f, hh = 0.f;
#pragma unroll
    for (int j = 0; j < 4; ++j) {
        ll = fmaf(colL[j], LO[j], ll);
        lh = fmaf(colH[j], LO[j], lh);
        hl = fmaf(colL[j], HI[j], hl);
        hh = fmaf(colH[j], HI[j], hh);
    }
    long cs = (long)Ho * Wo;
    float* d = dst + (long)b * 4 * cs + (long)oy * Wo + ox;
    d[0]      = ll;
    d[cs]     = lh;
    d[2 * cs] = hl;
    d[3 * cs] = hh;
}

// ---------------------------------------------------------------------------
// Patch predictor: one wave32 per 16-patch tile.
//   async-copy w2 (32KB) to LDS (GLOBAL_LOAD_ASYNC_TO_LDS_B128, ASYNCcnt)
//   stage1 (fused): x(10) -> w1 -> BN1 -> ReLU -> A tile (16 x 512) in LDS
//   GEMM: 128x v_wmma_f32_16x16x4_f32, all operands from LDS
//   epilogue: BN2+ReLU in C layout, LDS transpose, 16->16->2 MLP, flow = prior+res
// ---------------------------------------------------------------------------
__global__ void __launch_bounds__(32)
patch_predictor(const float* __restrict__ f1, const float* __restrict__ f2, int Hl,
                const float* __restrict__ flowNext,
                float* __restrict__ flowOut, int PH, int PW,
                const float* __restrict__ w1,  const float* __restrict__ b1,
                const float* __restrict__ bn1s, const float* __restrict__ bn1b,
                const float* __restrict__ bn1m, const float* __restrict__ bn1v,
                const float* __restrict__ w2,
                const float* __restrict__ b2,  const float* __restrict__ bn2s,
                const float* __restrict__ bn2b, const float* __restrict__ bn2m,
                const float* __restrict__ bn2v,
                const float* __restrict__ mw,  const float* __restrict__ mb,
                const float* __restrict__ oww, const float* __restrict__ ob,
                int coarsest)
{
    __shared__ alignas(16) float w2l[512 * 16];  // B matrix, row-major [k][n]
    __shared__ float Alds[16 * LDA];   // stage-1 activations, padded rows
    __shared__ float w1l[80];          // w1 (10 x 8)
    __shared__ float sc1[8], bi1[8];   // folded BN1 (b1 folded in)
    __shared__ float Hlds[16 * LDH];   // stage-2 activations (16 patches x 16)
    __shared__ float priorS[32];       // per-patch prior flow (y then x)

    const int lane = threadIdx.x;
    const long base = (long)blockIdx.x * 16;        // first patch id of tile
    const int PHPW = PH * PW;
    const int b   = (int)(base / PHPW);
    const int r0  = (int)(base % PHPW);
    const int py  = r0 / PW;
    const int px0 = r0 % PW;                        // PW is a multiple of 16

    // ---- stage w2 into LDS: async DMA path on CDNA5 (ASYNCcnt) ----
#if USE_ASYNC_LDS
#pragma unroll 4
    for (int i = lane; i < 2048; i += 32) {         // 2048 x b128 = 32KB
        __builtin_amdgcn_global_load_async_to_lds_b128(
            (v4i*)(w2 + (size_t)i * 4), (v4i*)(w2l + (size_t)i * 4), 0, 0);
    }
#else
    for (int i = lane; i < 8192; i += 32) w2l[i] = w2[i];
#endif

    // warm caches for the tiny MLP tables used in the epilogue
    __builtin_prefetch(mw, 0, 1);

    // ---- preload tiny params / fold BN1 ----
    for (int i = lane; i < 80; i += 32) w1l[i] = w1[i];
    if (lane < 8) {
        float sc = bn1s[lane] * rsqrtf(bn1v[lane] + EPSV);
        sc1[lane] = sc;
        bi1[lane] = (b1[lane] - bn1m[lane]) * sc + bn1b[lane];
    }
    // ---- per-patch prior flow (2x upsample of next-coarser flow) ----
    if (lane < 16) {
        float pry = 0.f, prx = 0.f;
        if (!coarsest) {
            int PHn = PH >> 1, PWn = PW >> 1;
            int px = px0 + lane;
            long o = (((long)b * 2) * PHn + (py >> 1)) * PWn + (px >> 1);
            pry = 2.f * flowNext[o];
            prx = 2.f * flowNext[o + (long)PHn * PWn];
        }
        priorS[lane]      = pry;
        priorS[16 + lane] = prx;
    }
    __syncthreads();

    // ---- stage 1: 16 patches x 64 pixels, 10 -> 8 features into A tile ----
    const long HW = (long)Hl * Hl;
    for (int t = lane; t < 1024; t += 32) {
        const int p   = t >> 6;          // patch in tile
        const int pix = t & 63;
        const int h   = pix >> 3;
        const int w   = pix & 7;
        const int px  = px0 + p;

        const float pry = priorS[p], prx = priorS[16 + p];
        const float wy = rintf(pry), wx = rintf(prx);   // round-half-even
        const float ry = pry - wy,  rx = prx - wx;
        int s0 = py * PP + (int)wy;                     // dynamic_slice clamp
        s0 = s0 < 0 ? 0 : (s0 > Hl - PP ? Hl - PP : s0);
        int s1 = px * PP + (int)wx;
        s1 = s1 < 0 ? 0 : (s1 > Hl - PP ? Hl - PP : s1);

        float x[10];
        const long o1 = (long)b * 4 * HW + (long)(py * PP + h) * Hl + (px * PP + w);
        x[0] = f1[o1];
        x[1] = f1[o1 + HW];
        x[2] = f1[o1 + 2 * HW];
        x[3] = f1[o1 + 3 * HW];
        const long o2 = (long)b * 4 * HW + (long)(s0 + h) * Hl + (s1 + w);
        x[4] = f2[o2];
        x[5] = f2[o2 + HW];
        x[6] = f2[o2 + 2 * HW];
        x[7] = f2[o2 + 3 * HW];
        x[8] = ry;
        x[9] = rx;

        float* arow = &Alds[p * LDA + pix * 8];
#pragma unroll
        for (int o = 0; o < 8; ++o) {
            float acc = 0.f;
#pragma unroll
            for (int c = 0; c < 10; ++c) acc = fmaf(x[c], w1l[c * 8 + o], acc);
            acc = fmaf(acc, sc1[o], bi1[o]);
            arow[o] = acc > 0.f ? acc : 0.f;
        }
    }
#if USE_ASYNC_LDS
    wait_async0();                       // w2 DMA complete before DS reads
#endif
    __syncthreads();

    // ---- GEMM: [16 x 512] @ [512 x 16] via V_WMMA_F32_16X16X4_F32 ----
    // A frag (ISA 16x4 f32): lanes 0-15 hold M=lane, K={k0,k0+1};
    //                        lanes 16-31 hold M=lane-16, K={k0+2,k0+3}.
    // B frag mirrors K split: VGPR0 = B[K=k0(+2)][N=lane&15], VGPR1 = K+1.
    const int half = lane >> 4;
    const int l    = lane & 15;
    v8f acc = {};
#pragma unroll 4
    for (int kk = 0; kk < 128; ++kk) {
        const int koff = kk * 4 + half * 2;
        v2f a = *(const v2f*)&Alds[l * LDA + koff];     // ds_load_b64
        v2f bb;
        bb.x = w2l[koff * 16 + l];                      // conflict-free banks
        bb.y = w2l[(koff + 1) * 16 + l];
        acc = __builtin_amdgcn_wmma_f32_16x16x4_f32(
                  false, a, false, bb, (short)0, acc, false, false);
    }

    // ---- BN2 + ReLU in C layout (VGPR r: M=r / M=8+r, N=lane&15) ----
    {
        const int n  = l;
        const float sc2 = bn2s[n] * rsqrtf(bn2v[n] + EPSV);
        const float bi2 = (b2[n] - bn2m[n]) * sc2 + bn2b[n];
#pragma unroll
        for (int r = 0; r < 8; ++r) {
            const int m = r + half * 8;
            float v = fmaf(acc[r], sc2, bi2);
            Hlds[m * LDH + n] = v > 0.f ? v : 0.f;
        }
    }
    __syncthreads();

    // ---- tiny MLP 16 -> 16 -> 2 per patch; flow = prior + residual ----
    if (lane < 16) {
        const int p  = lane;
        const int px = px0 + p;
        float hrow[16];
#pragma unroll
        for (int f = 0; f < 16; ++f) hrow[f] = Hlds[p * LDH + f];
        float m[16];
#pragma unroll
        for (int o = 0; o < 16; ++o) {
            float a2 = mb[o];
#pragma unroll
            for (int f = 0; f < 16; ++f) a2 = fmaf(hrow[f], mw[f * 16 + o], a2);
            m[o] = a2 > 0.f ? a2 : 0.f;
        }
        float ry = ob[0], rx = ob[1];
#pragma unroll
        for (int f = 0; f < 16; ++f) {
            ry = fmaf(m[f], oww[f * 2],     ry);
            rx = fmaf(m[f], oww[f * 2 + 1], rx);
        }
        const long o = (((long)b * 2) * PH + py) * PW + px;
        flowOut[o]                     = priorS[p]      + ry;
        flowOut[o + (long)PH * PW]     = priorS[16 + p] + rx;
    }
}

// ---------------------------------------------------------------------------
extern "C" void kernel_launch(void* const* d_in, const int* in_sizes, int n_in,
                              void* d_out, int out_size, void* d_ws, size_t ws_size,
                              hipStream_t stream)
{
    const float* img1 = (const float*)d_in[0];
    const float* img2 = (const float*)d_in[1];
    const float* w1   = (const float*)d_in[2];
    const float* b1   = (const float*)d_in[3];
    const float* bn1s = (const float*)d_in[4];
    const float* bn1b = (const float*)d_in[5];
    const float* bn1m = (const float*)d_in[6];
    const float* bn1v = (const float*)d_in[7];
    const float* w2   = (const float*)d_in[8];
    const float* b2   = (const float*)d_in[9];
    const float* bn2s = (const float*)d_in[10];
    const float* bn2b = (const float*)d_in[11];
    const float* bn2m = (const float*)d_in[12];
    const float* bn2v = (const float*)d_in[13];
    const float* mw   = (const float*)d_in[14];
    const float* mb   = (const float*)d_in[15];
    const float* ow   = (const float*)d_in[16];
    const float* ob   = (const float*)d_in[17];

    const int B = 16;
    const int Hl[3] = {512, 256, 128};

    // workspace layout: pyramids (both images, 3 levels) + coarse flow buffers
    char* ws = (char*)d_ws;
    size_t off = 0;
    float* pyr1[3]; float* pyr2[3];
    for (int lv = 0; lv < 3; ++lv) {
        size_t sz = (size_t)B * 4 * Hl[lv] * Hl[lv] * sizeof(float);
        pyr1[lv] = (float*)(ws + off); off += sz;
        pyr2[lv] = (float*)(ws + off); off += sz;
    }
    float* flowbuf[3];
    flowbuf[0] = (float*)d_out;                 // finest flow (B,2,64,64)
    for (int lv = 1; lv < 3; ++lv) {
        int PH = Hl[lv] / PP;
        size_t sz = (size_t)B * 2 * PH * PH * sizeof(float);
        flowbuf[lv] = (float*)(ws + off); off += sz;
    }

    // ---- wavelet pyramids ----
    for (int lv = 0; lv < 3; ++lv) {
        const int Hin = (lv == 0) ? 1024 : Hl[lv - 1];
        const long N = (long)B * Hl[lv] * Hl[lv];
        const int blocks = (int)((N + 255) / 256);
        const float* s1 = (lv == 0) ? img1 : pyr1[lv - 1];
        const float* s2 = (lv == 0) ? img2 : pyr2[lv - 1];
        const long bstride = (lv == 0) ? (long)1024 * 1024 : (long)4 * Hin * Hin;
        wavelet_down<<<blocks, 256, 0, stream>>>(s1, pyr1[lv], B, Hin, Hin, bstride);
        wavelet_down<<<blocks, 256, 0, stream>>>(s2, pyr2[lv], B, Hin, Hin, bstride);
    }

    // ---- coarse-to-fine predictor ----
    for (int lv = 2; lv >= 0; --lv) {
        const int PH = Hl[lv] / PP;
        const long Np = (long)B * PH * PH;        // multiple of 16
        const int ntiles = (int)(Np / 16);
        const float* fn = (lv == 2) ? flowbuf[0] /*unused*/ : flowbuf[lv + 1];
        patch_predictor<<<ntiles, 32, 0, stream>>>(
            pyr1[lv], pyr2[lv], Hl[lv],
            fn, flowbuf[lv], PH, PH,
            w1 + (size_t)lv * 80,  b1 + lv * 8,
            bn1s + lv * 8, bn1b + lv * 8, bn1m + lv * 8, bn1v + lv * 8,
            w2 + (size_t)lv * 8192,
            b2 + lv * 16, bn2s + lv * 16, bn2b + lv * 16, bn2m + lv * 16, bn2v + lv * 16,
            mw + (size_t)lv * 256, mb + lv * 16,
            ow + (size_t)lv * 32,  ob + lv * 2,
            (lv == 2) ? 1 : 0);
    }
}